// QCNN_46368466928137
// MI455X (gfx1250) — compile-verified
//
#include <hip/hip_runtime.h>
#include <math.h>

typedef float v2f __attribute__((ext_vector_type(2)));
typedef float v8f __attribute__((ext_vector_type(8)));

#define NREG 8  // 8 complex amplitudes (v2f) per lane; 32 lanes * 8 = 256-dim state

// XOR lane-permute via ds_swizzle_b32 (group-of-32 mode: and=0x1F, or=0, xor=XM)
template <int XM>
__device__ __forceinline__ float swz(float x) {
  return __int_as_float(
      __builtin_amdgcn_ds_swizzle(__float_as_int(x), (XM << 10) | 0x1F));
}
__device__ __forceinline__ float bperm(int byteIdx, float x) {
  return __int_as_float(__builtin_amdgcn_ds_bpermute(byteIdx, __float_as_int(x)));
}

// psi' = c*psi - i*s*psi_perm:  re' = c*re + s*im_p ; im' = c*im - s*re_p
// Packed form: ps' = ps*(c,c) + (im_p, re_p)*(s,-s)   -> v_pk_mul + v_pk_fma

__device__ __forceinline__ void rxx_reg(v2f cc, v2f sv, v2f* ps, const int km) {
  v2f t[NREG];
#pragma unroll
  for (int k = 0; k < NREG; ++k) { t[k].x = ps[k ^ km].y; t[k].y = ps[k ^ km].x; }
#pragma unroll
  for (int k = 0; k < NREG; ++k) ps[k] = ps[k] * cc + t[k] * sv;
}

template <int LM>
__device__ __forceinline__ void rxx_lane(v2f cc, v2f sv, v2f* ps) {
  v2f t[NREG];
#pragma unroll
  for (int k = 0; k < NREG; ++k) {
    t[k].x = swz<LM>(ps[k].y);   // im_p
    t[k].y = swz<LM>(ps[k].x);   // re_p
  }
#pragma unroll
  for (int k = 0; k < NREG; ++k) ps[k] = ps[k] * cc + t[k] * sv;
}

template <int LM>
__device__ __forceinline__ void rxx_mix(v2f cc, v2f sv, v2f* ps, const int km) {
  v2f t[NREG];
#pragma unroll
  for (int k = 0; k < NREG; ++k) {
    t[k].x = swz<LM>(ps[k ^ km].y);
    t[k].y = swz<LM>(ps[k ^ km].x);
  }
#pragma unroll
  for (int k = 0; k < NREG; ++k) ps[k] = ps[k] * cc + t[k] * sv;
}

// branch-free tanh / sigmoid on the v_exp/v_rcp fast path
__device__ __forceinline__ float fast_tanh(float x) {
  float e = __expf(2.0f * x);
  float r = __builtin_amdgcn_rcpf(e + 1.0f);
  return fmaf(-2.0f, r, 1.0f);
}
__device__ __forceinline__ float fast_sigmoid(float x) {
  float e = __expf(-x);
  return __builtin_amdgcn_rcpf(e + 1.0f);
}

__global__ void __launch_bounds__(256) qcnn_sim_kernel(
    const float* __restrict__ theta, const float* __restrict__ phi,
    const float* __restrict__ W1, const float* __restrict__ b1,
    const float* __restrict__ W2, const float* __restrict__ b2,
    float* __restrict__ out, int B)
{
  const int lane   = threadIdx.x & 31;
  const int wave   = blockIdx.x * (blockDim.x >> 5) + (threadIdx.x >> 5);
  const int nwaves = gridDim.x * (blockDim.x >> 5);
  const int m  = lane & 15;   // WMMA M index
  const int hv = lane >> 4;   // lane half (selects K offset / D row half)

  // --- Preload W1^T as two 16x4 WMMA A-tiles. A layout: M=lane%16, K = v + 2*(lane/16).
  v2f A0, A1;
  A0.x = W1[(2 * hv + 0) * 16 + m];
  A0.y = W1[(2 * hv + 1) * 16 + m];
  A1.x = W1[(2 * hv + 4) * 16 + m];
  A1.y = W1[(2 * hv + 5) * 16 + m];

  // Per-lane-half slice of b1 / W2 (rows hv*8 .. hv*8+7) matching D's row layout.
  float b1r[8], w2r[8];
#pragma unroll
  for (int r = 0; r < 8; ++r) { b1r[r] = b1[hv * 8 + r]; w2r[r] = W2[hv * 8 + r]; }
  const float bias2 = b2[0];

  const int pcl = __popc(lane);
  // Gather byte-indices for Walsh coefficients (precomputed, per lane half):
  //  hv=0 needs W8 (ev4), W4 (ev5);  hv=1 needs W16 (ev3), W2 (ev6), W1 (ev7)
  const int gi1 = (hv ? 16 : 8) << 2;
  const int gi2 = (hv ? 2 : 4) << 2;
  const int gi3 = (hv ? 1 : 4) << 2;   // hv=0 slot unused

  for (int b = wave; b < B; b += nwaves) {
    const float th = theta[b], ph = phi[b];
    float s, c;
    __sincosf(th, &s, &c);
    const v2f cc = {c, c};
    const v2f sv = {s, -s};

    // RZ diagonal phases: angle(idx) = phi*(2*popcount(idx)-8), idx=(k<<5)|lane.
    // Only 4 distinct per lane (popc(k) in 0..3): sincos(base), sincos(2phi), chain rotations.
    float cpc[4], spc[4];
    {
      float s2, c2, sb, cb;
      __sincosf(2.0f * ph, &s2, &c2);
      __sincosf(ph * (float)(2 * pcl - 8), &sb, &cb);
      cpc[0] = cb; spc[0] = sb;
#pragma unroll
      for (int j = 1; j < 4; ++j) {
        cpc[j] = cpc[j - 1] * c2 - spc[j - 1] * s2;
        spc[j] = cpc[j - 1] * s2 + spc[j - 1] * c2;
      }
    }

    // |++..+> scaled by first-cycle RZ phase (folded into init): psi = (1/16)*e^{i*ang}
    v2f ps[NREG];
#pragma unroll
    for (int k = 0; k < NREG; ++k) {
      const int pk = __builtin_popcount(k);  // compile-time per unrolled k
      ps[k].x = 0.0625f * cpc[pk];
      ps[k].y = 0.0625f * spc[pk];
    }

#pragma unroll
    for (int cyc = 0; cyc < 4; ++cyc) {
      // pairs (0,1),(2,3),(4,5),(6,7),(1,2),(3,4),(5,6)  (MSB-first bit mapping)
      rxx_reg(cc, sv, ps, 6);        // mask 0xC0 -> k^6 (register rename)
      rxx_mix<16>(cc, sv, ps, 1);    // mask 0x30 -> k^1, lane^16
      rxx_lane<12>(cc, sv, ps);      // mask 0x0C -> lane^12
      rxx_lane<3>(cc, sv, ps);       // mask 0x03 -> lane^3
      rxx_reg(cc, sv, ps, 3);        // mask 0x60 -> k^3 (register rename)
      rxx_lane<24>(cc, sv, ps);      // mask 0x18 -> lane^24
      rxx_lane<6>(cc, sv, ps);       // mask 0x06 -> lane^6
      if (cyc != 3) {                // RZ phase for the next cycle
#pragma unroll
        for (int k = 0; k < NREG; ++k) {
          const int pk = __builtin_popcount(k);
          const v2f cp = {cpc[pk], cpc[pk]};
          const v2f sp = {-spc[pk], spc[pk]};
          v2f sw = {ps[k].y, ps[k].x};           // folds into op_sel
          ps[k] = ps[k] * cp + sw * sp;
        }
      }
    }

    // probabilities and per-lane partial Z expectations
    float p[NREG], S = 0.0f;
#pragma unroll
    for (int k = 0; k < NREG; ++k) {
      p[k] = fmaf(ps[k].x, ps[k].x, ps[k].y * ps[k].y);
      S += p[k];
    }

    // qubits 0..2 live in k bits 2..0 (signs fold at compile time) -> plain butterflies
    float e0 = 0.0f, e1 = 0.0f, e2 = 0.0f;
#pragma unroll
    for (int k = 0; k < NREG; ++k) {
      e0 += ((k >> 2) & 1) ? -p[k] : p[k];
      e1 += ((k >> 1) & 1) ? -p[k] : p[k];
      e2 += ((k >> 0) & 1) ? -p[k] : p[k];
    }
#pragma unroll
    for (int sh = 16; sh >= 1; sh >>= 1) {
      // (templated swz needs constant; unrolled loop keeps sh constant)
      if (sh == 16) { e0 += swz<16>(e0); e1 += swz<16>(e1); e2 += swz<16>(e2); }
      if (sh == 8)  { e0 += swz<8>(e0);  e1 += swz<8>(e1);  e2 += swz<8>(e2);  }
      if (sh == 4)  { e0 += swz<4>(e0);  e1 += swz<4>(e1);  e2 += swz<4>(e2);  }
      if (sh == 2)  { e0 += swz<2>(e0);  e1 += swz<2>(e1);  e2 += swz<2>(e2);  }
      if (sh == 1)  { e0 += swz<1>(e0);  e1 += swz<1>(e1);  e2 += swz<1>(e2);  }
    }

    // qubits 3..7: ev[3+j] = Walsh coefficient W[2^(4-j)] of S over lanes.
    // Signed butterfly: after 5 steps lane m holds W[m].
    float w = S;
    {
      float pv;
      pv = swz<1>(w);  w = (lane & 1)  ? (pv - w) : (w + pv);
      pv = swz<2>(w);  w = (lane & 2)  ? (pv - w) : (w + pv);
      pv = swz<4>(w);  w = (lane & 4)  ? (pv - w) : (w + pv);
      pv = swz<8>(w);  w = (lane & 8)  ? (pv - w) : (w + pv);
      pv = swz<16>(w); w = (lane & 16) ? (pv - w) : (w + pv);
    }
    // Fetch the needed Walsh coefficients (3 bpermutes, indices precomputed):
    const float g1 = bperm(gi1, w);  // hv=0: W8=ev4 ; hv=1: W16=ev3
    const float g2 = bperm(gi2, w);  // hv=0: W4=ev5 ; hv=1: W2 =ev6
    const float g3 = bperm(gi3, w);  // hv=1: W1=ev7 (hv=0 unused)

    // --- MLP layer 1 via WMMA: D = W1^T(16x8) @ ev, ev broadcast across all N columns.
    // B layout: K = v + 2*(lane/16), N = lane%16 -> lane needs ev[2hv],ev[2hv+1],ev[2hv+4],ev[2hv+5]
    v2f Bt0, Bt1;
    Bt0.x = hv ? e2 : e0;
    Bt0.y = hv ? g1 : e1;   // ev3 : ev1
    Bt1.x = hv ? g2 : g1;   // ev6 : ev4
    Bt1.y = hv ? g3 : g2;   // ev7 : ev5
    v8f acc = {};
    acc = __builtin_amdgcn_wmma_f32_16x16x4_f32(false, A0, false, Bt0, (short)0, acc, false, false);
    acc = __builtin_amdgcn_wmma_f32_16x16x4_f32(false, A1, false, Bt1, (short)0, acc, false, false);

    // D row r (+8 for lane>=16) = hidden pre-activation; identical for every N column.
    float t = 0.0f;
#pragma unroll
    for (int r = 0; r < 8; ++r) t += fast_tanh(acc[r] + b1r[r]) * w2r[r];
    t += swz<16>(t);  // combine the two row-halves

    if (lane == 0) {
      out[b] = fast_sigmoid(t + bias2);
    }
  }
}

extern "C" void kernel_launch(void* const* d_in, const int* in_sizes, int n_in,
                              void* d_out, int out_size, void* d_ws, size_t ws_size,
                              hipStream_t stream) {
  (void)n_in; (void)d_ws; (void)ws_size; (void)out_size;
  const float* theta = (const float*)d_in[0];
  const float* phi   = (const float*)d_in[1];
  const float* W1    = (const float*)d_in[2];
  const float* b1    = (const float*)d_in[3];
  const float* W2    = (const float*)d_in[4];
  const float* b2    = (const float*)d_in[5];
  float* out = (float*)d_out;

  const int B = in_sizes[0];
  const int threads = 256;                 // 8 wave32s per workgroup
  const int wpb = threads / 32;
  int blocks = (B + wpb - 1) / wpb;        // 1 batch element per wave
  if (blocks > 8192) blocks = 8192;        // grid-stride covers the rest
  if (blocks < 1) blocks = 1;

  qcnn_sim_kernel<<<blocks, threads, 0, stream>>>(theta, phi, W1, b1, W2, b2, out, B);
}